// SSMProcessor_4209067950267
// MI455X (gfx1250) — compile-verified
//
#include <hip/hip_runtime.h>
#include <hip/hip_bf16.h>
#include <math.h>

typedef __attribute__((ext_vector_type(2))) float v2f;
typedef __attribute__((ext_vector_type(8))) float v8f;

#define BT   8
#define TLEN 4096
#define FDIM 512
#define HDIM 1024
#define NLAY 4
#define NC   64            // chunks along T
#define TC   (TLEN / NC)   // 64 steps per chunk
#define NCH  (BT * HDIM)   // 8192 channels

#define BSTRIDE 34         // LDS stride for transposed B tile: Bs[n*34 + k]

// ============================================================
// Kernel 1: GEMM  Y[m,n] = sum_k X[m,k] * W[k,n] + bias[n]
//   M=32768 (b*t), K=512, N=1024, f32, via V_WMMA_F32_16X16X4_F32
//   block = 256 thr = 8 waves; wave tile = 16(M) x 64(N) (4 wmma tiles)
//   B tile (32K x 64N) staged transposed into LDS by
//   GLOBAL_LOAD_ASYNC_TO_LDS_B32 (ASYNCcnt), double-buffered.
// ============================================================

// Issue 8 async B32 copies per thread: W[k0+k][n0+n] -> Bs[n*34 + k].
// Global side: consecutive lanes -> consecutive n (coalesced 128B/wave).
// LDS side: bank = (34n + k) mod 64, distinct across the wave (conflict-free).
__device__ __forceinline__ void stage_b_async(const float* __restrict__ W,
                                              const float* buf, int k0, int n0,
                                              int tid) {
  const unsigned base = (unsigned)(size_t)buf;  // low 32 bits = LDS byte offset
#pragma unroll
  for (int i = 0; i < 8; ++i) {
    const int flat = tid + 256 * i;   // 2048 elements
    const int n = flat & 63;
    const int k = flat >> 6;
    const unsigned lds_off = base + (unsigned)((n * BSTRIDE + k) * 4);
    const float* gp = W + (size_t)(k0 + k) * HDIM + n0 + n;
    asm volatile("global_load_async_to_lds_b32 %0, %1, off"
                 :: "v"(lds_off), "v"(gp) : "memory");
  }
}

__global__ __launch_bounds__(256) void gemm_wmma_f32(
    const float* __restrict__ X, const float* __restrict__ W,
    const float* __restrict__ bias, float* __restrict__ Y) {
  __shared__ float Bs[2][64 * BSTRIDE];

  const int tid  = threadIdx.x;
  const int lane = tid & 31;
  const int wave = tid >> 5;
  const int half = lane >> 4;     // lane-half selects K pair (A/B) / M half (C)
  const int l15  = lane & 15;

  const int m0 = (blockIdx.x * 8 + wave) * 16;
  const int n0 = blockIdx.y * 64;

  v8f acc0 = {}, acc1 = {}, acc2 = {}, acc3 = {};

  const float* xrow = X + (size_t)(m0 + l15) * FDIM;

  // pre-issue chunk 0 into buffer 0
  stage_b_async(W, &Bs[0][0], 0, n0, tid);

  for (int chunk = 0; chunk < FDIM / 32; ++chunk) {
    const int k0  = chunk * 32;
    const int cur = chunk & 1;

    // own batch done, then make everyone's batch visible
    asm volatile("s_wait_asynccnt 0x0" ::: "memory");
    __syncthreads();

    // overlap: fetch next chunk into the other buffer while computing
    if (chunk + 1 < FDIM / 32)
      stage_b_async(W, &Bs[cur ^ 1][0], k0 + 32, n0, tid);

    const float* bb = &Bs[cur][0];
#pragma unroll
    for (int kb = 0; kb < 32; kb += 4) {
      const int krow = kb + 2 * half;  // this lane-half's K pair (even)
      // A frag: 16x4 f32; lane holds row l15, K = krow, krow+1 (contiguous b64)
      v2f a = *(const v2f*)(xrow + k0 + krow);
      // B frags: transposed LDS -> K pair contiguous, one ds_load_b64 per tile
      v2f b0 = *(const v2f*)(bb + ( 0 + l15) * BSTRIDE + krow);
      v2f b1 = *(const v2f*)(bb + (16 + l15) * BSTRIDE + krow);
      v2f b2 = *(const v2f*)(bb + (32 + l15) * BSTRIDE + krow);
      v2f b3 = *(const v2f*)(bb + (48 + l15) * BSTRIDE + krow);

      acc0 = __builtin_amdgcn_wmma_f32_16x16x4_f32(false, a, false, b0, (short)0, acc0, false, false);
      acc1 = __builtin_amdgcn_wmma_f32_16x16x4_f32(false, a, false, b1, (short)0, acc1, false, false);
      acc2 = __builtin_amdgcn_wmma_f32_16x16x4_f32(false, a, false, b2, (short)0, acc2, false, false);
      acc3 = __builtin_amdgcn_wmma_f32_16x16x4_f32(false, a, false, b3, (short)0, acc3, false, false);
    }
  }

  // epilogue: C/D layout: VGPR i, half h -> M = i + 8h ; N = l15
  float bc0 = bias[n0 +  0 + l15];
  float bc1 = bias[n0 + 16 + l15];
  float bc2 = bias[n0 + 32 + l15];
  float bc3 = bias[n0 + 48 + l15];
#pragma unroll
  for (int i = 0; i < 8; ++i) {
    size_t row = (size_t)(m0 + i + 8 * half) * HDIM + n0;
    Y[row +  0 + l15] = acc0[i] + bc0;
    Y[row + 16 + l15] = acc1[i] + bc1;
    Y[row + 32 + l15] = acc2[i] + bc2;
    Y[row + 48 + l15] = acc3[i] + bc3;
  }
}

// ============================================================
// Fused 4-layer SSM stack = affine recurrence sigma' = M*sigma + v*u + c
// M is 4x4 lower-triangular, per-h only. Chunked scan over T.
// ============================================================

__device__ inline void mat4mul(const float* X, const float* Yv, float* Z) {
#pragma unroll
  for (int i = 0; i < 4; ++i)
#pragma unroll
    for (int j = 0; j < 4; ++j) {
      float acc = 0.f;
#pragma unroll
      for (int k = 0; k < 4; ++k) acc += X[i * 4 + k] * Yv[k * 4 + j];
      Z[i * 4 + j] = acc;
    }
}

__global__ void ssm_precompute(const float* __restrict__ A,
                               const float* __restrict__ Bw,
                               const float* __restrict__ Cw,
                               float* __restrict__ Mp) {
  int h = blockIdx.x * blockDim.x + threadIdx.x;
  if (h >= HDIM) return;
  float a[4], Bv[4], Cv[4];
#pragma unroll
  for (int l = 0; l < 4; ++l) {
    a[l]  = 1.f / (1.f + expf(-A[l * HDIM + h]));
    Bv[l] = Bw[l * HDIM + h];
    Cv[l] = Cw[l * HDIM + h];
  }
  // Build M column-by-column: one linear step (u=0, bias=0) from e_j
  float M[16];
#pragma unroll
  for (int j = 0; j < 4; ++j) {
    float s[4] = {0.f, 0.f, 0.f, 0.f};
    s[j] = 1.f;
    float ns[4], y;
    ns[0] = a[0] * s[0];               y = Cv[0] * ns[0];
    ns[1] = a[1] * s[1] + Bv[1] * y;   y = Cv[1] * ns[1];
    ns[2] = a[2] * s[2] + Bv[2] * y;   y = Cv[2] * ns[2];
    ns[3] = a[3] * s[3] + Bv[3] * y;
#pragma unroll
    for (int i = 0; i < 4; ++i) M[i * 4 + j] = ns[i];
  }
  // P = M^64 = six squarings
  float P[16], Tm[16];
#pragma unroll
  for (int i = 0; i < 16; ++i) P[i] = M[i];
#pragma unroll
  for (int it = 0; it < 6; ++it) {
    mat4mul(P, P, Tm);
#pragma unroll
    for (int i = 0; i < 16; ++i) P[i] = Tm[i];
  }
#pragma unroll
  for (int i = 0; i < 16; ++i) Mp[(size_t)h * 16 + i] = P[i];
}

// Pass 1 (FINAL=false): chunk-local scan from zero init, emit end state.
// Pass 3 (FINAL=true): scan with corrected init, write y in place over Y.
template <bool FINAL>
__global__ __launch_bounds__(256) void ssm_chunk(
    const float* __restrict__ A, const float* __restrict__ Bw,
    const float* __restrict__ Cw, const float* __restrict__ Dw,
    float* __restrict__ Y, const float4* __restrict__ sig_in,
    float4* __restrict__ sig_out) {
  int idx = blockIdx.x * 256 + threadIdx.x;  // NC * NCH threads
  int c   = idx >> 13;                       // chunk
  int rem = idx & (NCH - 1);                 // (b,h)
  int b   = rem >> 10;
  int h   = rem & (HDIM - 1);

  float a[4], Bv[4], Cv[4], Dv[4];
#pragma unroll
  for (int l = 0; l < 4; ++l) {
    a[l]  = 1.f / (1.f + expf(-A[l * HDIM + h]));
    Bv[l] = Bw[l * HDIM + h];
    Cv[l] = Cw[l * HDIM + h];
    Dv[l] = Dw[l * HDIM + h];
  }

  float s0 = 0.f, s1 = 0.f, s2 = 0.f, s3 = 0.f;
  if (FINAL) {
    float4 si = sig_in[(size_t)c * NCH + rem];
    s0 = si.x; s1 = si.y; s2 = si.z; s3 = si.w;
  }

  float* yp = Y + ((size_t)b * TLEN + (size_t)c * TC) * HDIM + h;
#pragma unroll 4
  for (int t = 0; t < TC; ++t) {
    float u = yp[(size_t)t * HDIM];
    s0 = a[0] * s0 + Bv[0] * u;  float y = Cv[0] * s0 + Dv[0];
    s1 = a[1] * s1 + Bv[1] * y;  y = Cv[1] * s1 + Dv[1];
    s2 = a[2] * s2 + Bv[2] * y;  y = Cv[2] * s2 + Dv[2];
    s3 = a[3] * s3 + Bv[3] * y;  y = Cv[3] * s3 + Dv[3];
    if (FINAL) yp[(size_t)t * HDIM] = y;   // in-place: read-before-write per elem
  }
  if (!FINAL) sig_out[(size_t)c * NCH + rem] = make_float4(s0, s1, s2, s3);
}

// Pass 2: sequential combine across chunks (per channel), using Mp = M^TC.
__global__ void ssm_combine(const float* __restrict__ Mp,
                            const float4* __restrict__ sig_local,
                            float4* __restrict__ sig_start) {
  int idx = blockIdx.x * blockDim.x + threadIdx.x;  // NCH
  int h = idx & (HDIM - 1);
  float m[16];
#pragma unroll
  for (int i = 0; i < 16; ++i) m[i] = Mp[(size_t)h * 16 + i];
  float4 s = make_float4(0.f, 0.f, 0.f, 0.f);
  for (int c = 0; c < NC; ++c) {
    sig_start[(size_t)c * NCH + idx] = s;
    float4 lo = sig_local[(size_t)c * NCH + idx];
    float4 ns;
    ns.x = m[0]  * s.x + m[1]  * s.y + m[2]  * s.z + m[3]  * s.w + lo.x;
    ns.y = m[4]  * s.x + m[5]  * s.y + m[6]  * s.z + m[7]  * s.w + lo.y;
    ns.z = m[8]  * s.x + m[9]  * s.y + m[10] * s.z + m[11] * s.w + lo.z;
    ns.w = m[12] * s.x + m[13] * s.y + m[14] * s.z + m[15] * s.w + lo.w;
    s = ns;
  }
}

extern "C" void kernel_launch(void* const* d_in, const int* in_sizes, int n_in,
                              void* d_out, int out_size, void* d_ws, size_t ws_size,
                              hipStream_t stream) {
  (void)in_sizes; (void)n_in; (void)out_size; (void)ws_size;
  const float* x   = (const float*)d_in[0];  // [8,4096,512]
  const float* Win = (const float*)d_in[1];  // [512,1024]
  const float* bin = (const float*)d_in[2];  // [1024]
  const float* A   = (const float*)d_in[3];  // [4,1024]
  const float* Bw  = (const float*)d_in[4];
  const float* Cw  = (const float*)d_in[5];
  const float* Dw  = (const float*)d_in[6];
  float* out = (float*)d_out;                // [8,4096,1024] (y0 then y, in place)

  char* ws = (char*)d_ws;
  float*  Mp   = (float*)ws;                                   //  64 KiB
  float4* sigL = (float4*)(ws + (64 << 10));                   //   8 MiB
  float4* sigS = (float4*)(ws + (64 << 10) + (8 << 20));       //   8 MiB

  // 1) GEMM -> out (acts as y0)
  dim3 gg(32768 / (8 * 16), HDIM / 64);  // (256, 16)
  gemm_wmma_f32<<<gg, 256, 0, stream>>>(x, Win, bin, out);

  // 2) per-h transition matrix M^TC
  ssm_precompute<<<HDIM / 256, 256, 0, stream>>>(A, Bw, Cw, Mp);

  // 3) chunk-local scans (zero init)
  ssm_chunk<false><<<(NC * NCH) / 256, 256, 0, stream>>>(A, Bw, Cw, Dw, out, nullptr, sigL);

  // 4) sequential chunk combine
  ssm_combine<<<NCH / 256, 256, 0, stream>>>(Mp, sigL, sigS);

  // 5) final scans with corrected init, write y in place
  ssm_chunk<true><<<(NC * NCH) / 256, 256, 0, stream>>>(A, Bw, Cw, Dw, out, sigS, nullptr);
}